// OracleMoE_76965813944414
// MI455X (gfx1250) — compile-verified
//
#include <hip/hip_runtime.h>
#include <cmath>

typedef __attribute__((ext_vector_type(2))) float v2f;
typedef __attribute__((ext_vector_type(8))) float v8f;

// Problem constants (fixed by the reference).
constexpr int B = 8, N = 1024, D = 1024, E = 16, H = 4096;
constexpr int CAP = 80;                 // int(N * 1.25 / E), >= 4
constexpr int MT  = CAP / 16;           // 5 row tiles of 16
constexpr int BK  = 64;                 // K chunk staged in LDS
constexpr int LDS_STRIDE = BK + 4;      // pad to avoid 64-bank conflicts

__device__ __forceinline__ float gelu_exact(float x) {
    // gelu(approximate=False): 0.5*x*(1+erf(x/sqrt(2)))
    return 0.5f * x * (1.0f + erff(x * 0.70710678118654752f));
}

__global__ void zero_f4(float4* __restrict__ p, long n4) {
    long i = (long)blockIdx.x * blockDim.x + threadIdx.x;
    if (i < n4) p[i] = make_float4(0.f, 0.f, 0.f, 0.f);
}

// C[b] (rows 0..CAP-1, row stride ldn) = [gelu]( A[b] (CAP x K, row stride lda)
//                                               @ W[expert_b] (K x ldn, row-major) )
template<int WAVES, bool GELU>
__global__ __launch_bounds__(WAVES * 32)
void moe_gemm(const float* __restrict__ Abase,
              const float* __restrict__ Wbase,
              float*       __restrict__ Cbase,
              const int*   __restrict__ current_y,
              int K, int lda, int ldn,
              long strideA, long strideC)
{
    __shared__ float lds[CAP * LDS_STRIDE];

    const int b      = blockIdx.y;
    const int expert = ((unsigned)current_y[b]) % E;
    const int tileN0 = blockIdx.x * (WAVES * 16);

    const float* A = Abase + (long)b * strideA;
    const float* W = Wbase + (long)expert * (long)K * ldn;
    float*       C = Cbase + (long)b * strideC;

    const int tid    = threadIdx.x;
    const int wave   = tid >> 5;
    const int lane   = tid & 31;
    const int laneHi = lane >> 4;   // 0: lanes 0-15, 1: lanes 16-31
    const int laneLo = lane & 15;

    const int n0 = tileN0 + wave * 16;   // this wave's 16-wide output column strip

    v8f acc[MT] = {};                    // 5 x (16x16 f32) accumulators

    for (int k0 = 0; k0 < K; k0 += BK) {
        __syncthreads();
        // Cooperative, coalesced load of the CAP x BK activation chunk into LDS.
        for (int idx = tid; idx < CAP * BK; idx += WAVES * 32) {
            int r = idx / BK, c = idx % BK;
            lds[r * LDS_STRIDE + c] = A[(long)r * lda + (k0 + c)];
        }
        __syncthreads();

        #pragma unroll
        for (int kk = 0; kk < BK; kk += 4) {
            // B fragment (4x16): lanes 0-15 read rows kk,kk+1; lanes 16-31 rows kk+2,kk+3.
            // 16 contiguous floats per half-wave -> coalesced 64B segments from HBM.
            const long woff = (long)(k0 + kk + laneHi * 2) * ldn + (n0 + laneLo);
            v2f bfrag;
            bfrag.x = W[woff];
            bfrag.y = W[woff + ldn];

            #pragma unroll
            for (int m = 0; m < MT; ++m) {
                // A fragment (16x4) from LDS: lane maps to row, laneHi selects K pair.
                const float* ap = &lds[(m * 16 + laneLo) * LDS_STRIDE + kk + laneHi * 2];
                v2f afrag;
                afrag.x = ap[0];
                afrag.y = ap[1];
                acc[m] = __builtin_amdgcn_wmma_f32_16x16x4_f32(
                    false, afrag, false, bfrag, (short)0, acc[m], false, false);
            }
        }
    }

    // C/D layout: VGPR r -> (M = r + 8*laneHi + 16*m, N = n0 + laneLo)
    #pragma unroll
    for (int m = 0; m < MT; ++m) {
        #pragma unroll
        for (int r = 0; r < 8; ++r) {
            int row = m * 16 + r + laneHi * 8;
            float v = acc[m][r];
            if (GELU) v = gelu_exact(v);
            C[(long)row * ldn + (n0 + laneLo)] = v;
        }
    }
}

extern "C" void kernel_launch(void* const* d_in, const int* in_sizes, int n_in,
                              void* d_out, int out_size, void* d_ws, size_t ws_size,
                              hipStream_t stream) {
    const float* x   = (const float*)d_in[0];  // (B, N, D) fp32
    const int*   y   = (const int*)  d_in[1];  // (B,) int32
    const float* w1  = (const float*)d_in[2];  // (E, D, H) fp32
    const float* w2  = (const float*)d_in[3];  // (E, H, D) fp32
    float*       out = (float*)d_out;          // (B, N, D) fp32
    float*       hid = (float*)d_ws;           // (B, CAP, H) fp32 scratch (10.5 MB)

    // 1) zero the full output (tokens >= CAP must be zero).
    long n4 = (long)B * N * D / 4;
    zero_f4<<<dim3((unsigned)((n4 + 255) / 256)), 256, 0, stream>>>((float4*)out, n4);

    // 2) hidden = gelu(X[:, :CAP, :] @ W1[e_b]);  M=CAP, K=D, N=H
    dim3 g1(H / (8 * 16), B);
    moe_gemm<8, true><<<g1, 8 * 32, 0, stream>>>(
        x, w1, hid, y, /*K=*/D, /*lda=*/D, /*ldn=*/H,
        /*strideA=*/(long)N * D, /*strideC=*/(long)CAP * H);

    // 3) out[:, :CAP, :] = hidden @ W2[e_b];  M=CAP, K=H, N=D
    dim3 g2(D / (4 * 16), B);
    moe_gemm<4, false><<<g2, 4 * 32, 0, stream>>>(
        hid, w2, out, y, /*K=*/H, /*lda=*/H, /*ldn=*/D,
        /*strideA=*/(long)CAP * H, /*strideC=*/(long)N * D);
}